// Net_49065706389774
// MI455X (gfx1250) — compile-verified
//
#include <hip/hip_runtime.h>

// GCN: out = GCNConv2( relu(GCNConv1(x)) ) @ WL + bL  on MI455X (gfx1250).
// Strategy: L2-resident pipeline; fp32 WMMA (16x16x4) for all dense GEMMs;
// dinv pre-scaling so the per-edge hot loop is gather + atomicAdd only.
// Store epilogues use a wave-uniform fast path to avoid per-element EXEC churn.

typedef float v2f __attribute__((ext_vector_type(2)));
typedef float v8f __attribute__((ext_vector_type(8)));

constexpr int F_IN = 16, H1 = 64, H2 = 128, F_OUT = 16;

// ---------- degree / normalization ----------
__global__ void init_ones_kernel(float* __restrict__ deg, int N) {
  int i = blockIdx.x * blockDim.x + threadIdx.x;
  if (i < N) deg[i] = 1.0f;                       // self-loop contributes 1
}

__global__ void deg_kernel(const int* __restrict__ dst, float* __restrict__ deg, int E) {
  int i = blockIdx.x * blockDim.x + threadIdx.x;
  if (i < E) atomicAdd(&deg[dst[i]], 1.0f);       // global_atomic_add_f32
}

__global__ void rsqrt_kernel(float* __restrict__ dinv, int N) {
  int i = blockIdx.x * blockDim.x + threadIdx.x;
  if (i < N) dinv[i] = rsqrtf(dinv[i]);           // deg >= 1 always (self-loop)
}

// ---------- GEMM1: h1t = (x @ W1) * dinv[row]   [N,16]x[16,64] ----------
// block = 128 threads = 4 waves; wave w -> col tile w; block -> 16 rows.
__global__ void gemm1_kernel(const float* __restrict__ x, const float* __restrict__ W1,
                             const float* __restrict__ dinv, float* __restrict__ h1t, int N) {
  const int lane = threadIdx.x & 31;
  const int wave = threadIdx.x >> 5;              // 0..3 (col tile)
  const int hi = lane >> 4, lo = lane & 15;
  const int row0 = blockIdx.x << 4;
  const int col = (wave << 4) + lo;               // 0..63
  int am = row0 + lo; if (am > N - 1) am = N - 1; // clamped read row
  v8f c = {};
#pragma unroll
  for (int kk = 0; kk < 4; ++kk) {                // K = 16 = 4 x 4
    const int k = (kk << 2) + (hi << 1);
    v2f a, b;
    a.x = x[am * F_IN + k];
    a.y = x[am * F_IN + k + 1];
    b.x = W1[k * H1 + col];
    b.y = W1[(k + 1) * H1 + col];
    c = __builtin_amdgcn_wmma_f32_16x16x4_f32(false, a, false, b, (short)0, c, false, false);
  }
  if (row0 + 16 <= N) {                           // wave-uniform fast path
#pragma unroll
    for (int r = 0; r < 8; ++r) {
      const int row = row0 + (hi << 3) + r;
      h1t[(size_t)row * H1 + col] = c[r] * dinv[row];
    }
  } else {                                        // ragged tail (rare)
#pragma unroll
    for (int r = 0; r < 8; ++r) {
      const int row = row0 + (hi << 3) + r;
      if (row < N) h1t[(size_t)row * H1 + col] = c[r] * dinv[row];
    }
  }
}

// ---------- GEMM2: h2t = (h @ W2) * dinv[row]   [N,64]x[64,128] ----------
// block = 256 threads = 8 waves; wave w -> col tile w; block -> 16 rows.
__global__ void gemm2_kernel(const float* __restrict__ h, const float* __restrict__ W2,
                             const float* __restrict__ dinv, float* __restrict__ h2t, int N) {
  const int lane = threadIdx.x & 31;
  const int wave = threadIdx.x >> 5;              // 0..7 (col tile)
  const int hi = lane >> 4, lo = lane & 15;
  const int row0 = blockIdx.x << 4;
  const int col = (wave << 4) + lo;               // 0..127
  int am = row0 + lo; if (am > N - 1) am = N - 1;
  v8f c = {};
#pragma unroll 4
  for (int kk = 0; kk < 16; ++kk) {               // K = 64 = 16 x 4
    const int k = (kk << 2) + (hi << 1);
    v2f a, b;
    a.x = h[(size_t)am * H1 + k];
    a.y = h[(size_t)am * H1 + k + 1];
    b.x = W2[k * H2 + col];
    b.y = W2[(k + 1) * H2 + col];
    c = __builtin_amdgcn_wmma_f32_16x16x4_f32(false, a, false, b, (short)0, c, false, false);
  }
  if (row0 + 16 <= N) {                           // wave-uniform fast path
#pragma unroll
    for (int r = 0; r < 8; ++r) {
      const int row = row0 + (hi << 3) + r;
      h2t[(size_t)row * H2 + col] = c[r] * dinv[row];
    }
  } else {
#pragma unroll
    for (int r = 0; r < 8; ++r) {
      const int row = row0 + (hi << 3) + r;
      if (row < N) h2t[(size_t)row * H2 + col] = c[r] * dinv[row];
    }
  }
}

// ---------- GEMM3 (fused post2): out = (dinv*agg2 + b2) @ WL + bL ----------
// block = 256 threads = 8 waves; wave w -> row tile; cols = 16 = one tile.
__global__ void gemm3_kernel(const float* __restrict__ agg2, const float* __restrict__ WL,
                             const float* __restrict__ b2, const float* __restrict__ bL,
                             const float* __restrict__ dinv, float* __restrict__ out, int N) {
  const int lane = threadIdx.x & 31;
  const int wave = threadIdx.x >> 5;              // 0..7 (row tile)
  const int hi = lane >> 4, lo = lane & 15;
  const int row0 = (blockIdx.x << 7) + (wave << 4);
  if (row0 >= N) return;                          // wave-uniform -> EXEC stays full
  int am = row0 + lo; if (am > N - 1) am = N - 1;
  const float dv = dinv[am];
  v8f c = {};
#pragma unroll 4
  for (int kk = 0; kk < 32; ++kk) {               // K = 128 = 32 x 4
    const int k = (kk << 2) + (hi << 1);
    v2f a, b;
    a.x = dv * agg2[(size_t)am * H2 + k]     + b2[k];
    a.y = dv * agg2[(size_t)am * H2 + k + 1] + b2[k + 1];
    b.x = WL[k * F_OUT + lo];
    b.y = WL[(k + 1) * F_OUT + lo];
    c = __builtin_amdgcn_wmma_f32_16x16x4_f32(false, a, false, b, (short)0, c, false, false);
  }
  const float bias = bL[lo];
  if (row0 + 16 <= N) {                           // wave-uniform fast path
#pragma unroll
    for (int r = 0; r < 8; ++r) {
      const int row = row0 + (hi << 3) + r;
      out[(size_t)row * F_OUT + lo] = c[r] + bias;
    }
  } else {
#pragma unroll
    for (int r = 0; r < 8; ++r) {
      const int row = row0 + (hi << 3) + r;
      if (row < N) out[(size_t)row * F_OUT + lo] = c[r] + bias;
    }
  }
}

// ---------- edge scatter: agg[dst] += ht[src] (rows pre-scaled by dinv[src]) ----------
// 16 threads/edge, float4 gather (global_load_b128) + 4 fp32 atomics each.
__global__ void agg_kernel_64(const int* __restrict__ src, const int* __restrict__ dst,
                              const float* __restrict__ h, float* __restrict__ agg, int E) {
  const long long t = (long long)blockIdx.x * blockDim.x + threadIdx.x;
  if (t >= (long long)E * 16) return;
  const int e = (int)(t >> 4);
  const int f = ((int)t & 15) << 2;
  const int s = src[e], d = dst[e];
  const float4 v = *reinterpret_cast<const float4*>(h + (size_t)s * H1 + f);
  float* p = agg + (size_t)d * H1 + f;
  atomicAdd(p + 0, v.x); atomicAdd(p + 1, v.y);
  atomicAdd(p + 2, v.z); atomicAdd(p + 3, v.w);
}

// 32 threads/edge for the 128-wide layer.
__global__ void agg_kernel_128(const int* __restrict__ src, const int* __restrict__ dst,
                               const float* __restrict__ h, float* __restrict__ agg, int E) {
  const long long t = (long long)blockIdx.x * blockDim.x + threadIdx.x;
  if (t >= (long long)E * 32) return;
  const int e = (int)(t >> 5);
  const int f = ((int)t & 31) << 2;
  const int s = src[e], d = dst[e];
  const float4 v = *reinterpret_cast<const float4*>(h + (size_t)s * H2 + f);
  float* p = agg + (size_t)d * H2 + f;
  atomicAdd(p + 0, v.x); atomicAdd(p + 1, v.y);
  atomicAdd(p + 2, v.z); atomicAdd(p + 3, v.w);
}

// ---------- post layer 1: h = relu(dinv[row]*agg1 + b1), in place ----------
__global__ void post1_kernel(const float* __restrict__ b1, const float* __restrict__ dinv,
                             float* __restrict__ agg, int N) {
  const int i = blockIdx.x * blockDim.x + threadIdx.x;
  if (i >= N * H1) return;
  const int row = i >> 6;
  const int f = i & 63;
  const float v = dinv[row] * agg[i] + b1[f];
  agg[i] = fmaxf(v, 0.0f);
}

static inline unsigned cdiv(long long a, long long b) { return (unsigned)((a + b - 1) / b); }

extern "C" void kernel_launch(void* const* d_in, const int* in_sizes, int n_in,
                              void* d_out, int out_size, void* d_ws, size_t ws_size,
                              hipStream_t stream) {
  const float* x  = (const float*)d_in[0];
  const int*   ei = (const int*)d_in[1];
  const float* W1 = (const float*)d_in[2];
  const float* b1 = (const float*)d_in[3];
  const float* W2 = (const float*)d_in[4];
  const float* b2 = (const float*)d_in[5];
  const float* WL = (const float*)d_in[6];
  const float* bL = (const float*)d_in[7];
  float* out = (float*)d_out;

  const int N = in_sizes[0] / F_IN;
  const int E = in_sizes[1] / 2;
  const int* src = ei;
  const int* dst = ei + E;

  // Workspace layout (floats), 321*N total (~128 MB for N=100k; L2-resident working set):
  //   dinv[N] | bufA[64N] (agg1, then h in place) | bufB[256N]:
  //     h1t = bufB[0:64N] (dead after scatter-1); h2t = bufB[0:128N]; agg2 = bufB[128N:256N]
  float* ws = (float*)d_ws;
  const size_t nAl = ((size_t)N + 63) & ~(size_t)63;
  float* dinv = ws;
  float* bufA = ws + nAl;                   // 64N  : agg1 -> h
  float* bufB = bufA + 64 * nAl;            // 256N
  float* h1t  = bufB;                       // 64N
  float* h2t  = bufB;                       // 128N (reuses h1t region after it is dead)
  float* agg2 = bufB + 128 * nAl;           // 128N

  // 1) deg (with self-loop) -> dinv = rsqrt(deg), all in place
  init_ones_kernel<<<cdiv(N, 256), 256, 0, stream>>>(dinv, N);
  deg_kernel<<<cdiv(E, 256), 256, 0, stream>>>(dst, dinv, E);
  rsqrt_kernel<<<cdiv(N, 256), 256, 0, stream>>>(dinv, N);

  // 2) layer 1: h1t = (x@W1)*dinv ; agg1 = h1t (self-loop) ; scatter edges ; bias+relu
  gemm1_kernel<<<cdiv(N, 16), 128, 0, stream>>>(x, W1, dinv, h1t, N);
  hipMemcpyAsync(bufA, h1t, (size_t)N * H1 * sizeof(float), hipMemcpyDeviceToDevice, stream);
  agg_kernel_64<<<cdiv((long long)E * 16, 256), 256, 0, stream>>>(src, dst, h1t, bufA, E);
  post1_kernel<<<cdiv((long long)N * H1, 256), 256, 0, stream>>>(b1, dinv, bufA, N);

  // 3) layer 2: h2t = (h@W2)*dinv ; agg2 = h2t (self-loop) ; scatter edges
  gemm2_kernel<<<cdiv(N, 16), 256, 0, stream>>>(bufA, W2, dinv, h2t, N);
  hipMemcpyAsync(agg2, h2t, (size_t)N * H2 * sizeof(float), hipMemcpyDeviceToDevice, stream);
  agg_kernel_128<<<cdiv((long long)E * 32, 256), 256, 0, stream>>>(src, dst, h2t, agg2, E);

  // 4) head, fused with layer-2 normalization/bias: out = (dinv*agg2 + b2) @ WL + bL
  gemm3_kernel<<<cdiv(N, 128), 256, 0, stream>>>(agg2, WL, b2, bL, dinv, out, N);
}